// MultiHeadAttnBlock_35880156791648
// MI455X (gfx1250) — compile-verified
//
#include <hip/hip_runtime.h>

#define C 128
#define HW 4096
#define HEADS 4
#define CPH 32
#define NT 256                 // 16-wide tiles along hw
#define SCALE 0.08838834764831845f   // 128^-0.5 (full C, as in reference)

typedef __attribute__((ext_vector_type(2))) float v2f;
typedef __attribute__((ext_vector_type(8))) float v8f;

__device__ __forceinline__ v8f wmma_f32(v2f a, v2f b, v8f c) {
  // D = A(16x4) * B(4x16) + C, fp32, wave32
  return __builtin_amdgcn_wmma_f32_16x16x4_f32(false, a, false, b, (short)0, c, false, false);
}

// ---------------- GroupNorm ----------------
__global__ void gn_stats_kernel(const float* __restrict__ x,
                                float* __restrict__ gmean, float* __restrict__ grstd) {
  const int g = blockIdx.x;          // 32 groups, 4 channels x 4096 each
  const int tid = threadIdx.x;       // 256 threads
  const float* xp = x + (size_t)g * 4 * HW;
  float s = 0.f, s2 = 0.f;
  for (int i = tid; i < 4 * HW; i += 256) {
    float v = xp[i];
    s += v; s2 += v * v;
  }
  __shared__ float rs[256], rs2[256];
  rs[tid] = s; rs2[tid] = s2;
  __syncthreads();
  for (int off = 128; off > 0; off >>= 1) {
    if (tid < off) { rs[tid] += rs[tid + off]; rs2[tid] += rs2[tid + off]; }
    __syncthreads();
  }
  if (tid == 0) {
    const float inv_n = 1.0f / (4.0f * HW);
    float mu = rs[0] * inv_n;
    float var = rs2[0] * inv_n - mu * mu;
    gmean[g] = mu;
    grstd[g] = rsqrtf(var + 1e-6f);
  }
}

__global__ void gn_apply_kernel(const float* __restrict__ x,
                                const float* __restrict__ gamma, const float* __restrict__ beta,
                                const float* __restrict__ gmean, const float* __restrict__ grstd,
                                float* __restrict__ xn) {
  const int idx = blockIdx.x * 256 + threadIdx.x;   // C*HW total
  const int c = idx >> 12;     // /HW
  const int g = c >> 2;        // 4 channels per group
  xn[idx] = (x[idx] - gmean[g]) * grstd[g] * gamma[c] + beta[c];
}

// ---------------- 128x128 x (128xHW) GEMM (1x1 conv) ----------------
// mode 0: Y[o,p] = W@X + b      (row-major [o][p])
// mode 1: store V transposed:   vt[(head*HW + p)*CPH + c_idx], head=o&3, c_idx=o>>2
// mode 2: mode 0 + residual
__global__ void gemm_kernel(const float* __restrict__ W, const float* __restrict__ bias,
                            const float* __restrict__ X, float* __restrict__ Y,
                            const float* __restrict__ resid, int mode) {
  const int lane = threadIdx.x & 31;
  const int wave = threadIdx.x >> 5;   // 8 waves -> 8 o-tiles (covers all 128 outputs)
  const int m = lane & 15;
  const int hi = lane >> 4;
  const int o0 = wave * 16;
  const int p0 = blockIdx.x * 16;
  v8f acc = {};
  for (int ct = 0; ct < 32; ++ct) {
    const int c0 = ct * 4 + hi * 2;
    v2f a, b;
    a.x = W[(o0 + m) * C + c0];
    a.y = W[(o0 + m) * C + c0 + 1];
    b.x = X[c0 * HW + p0 + m];
    b.y = X[(c0 + 1) * HW + p0 + m];
    acc = wmma_f32(a, b, acc);
  }
#pragma unroll
  for (int r = 0; r < 8; ++r) {
    const int o = o0 + r + 8 * hi;
    const int p = p0 + m;
    float val = acc[r] + bias[o];
    if (mode == 2) val += resid[o * HW + p];
    if (mode == 1) {
      Y[((o & 3) * HW + p) * CPH + (o >> 2)] = val;
    } else {
      Y[o * HW + p] = val;
    }
  }
}

// ---------------- Pass 1: per-key softmax stats over the QUERY axis ----------------
__global__ void attn_stats_kernel(const float* __restrict__ q, const float* __restrict__ kmat,
                                  float* __restrict__ ms, float* __restrict__ ss) {
  const int lane = threadIdx.x & 31;
  const int wave = threadIdx.x >> 5;
  const int gw = blockIdx.x * 8 + wave;     // 1024 waves total
  const int head = gw >> 8;
  const int ktile = gw & 255;
  const int m = lane & 15;
  const int hi = lane >> 4;
  const int kcol = ktile * 16 + m;

  // K B-tile is invariant over the query loop: hoist (channels are head-strided by 4)
  v2f kb[8];
#pragma unroll
  for (int ct = 0; ct < 8; ++ct) {
    const int c0 = ct * 4 + hi * 2;
    kb[ct].x = kmat[(head + 4 * c0) * HW + kcol];
    kb[ct].y = kmat[(head + 4 * (c0 + 1)) * HW + kcol];
  }

  float runm = -3.4e38f, runs = 0.f;
  for (int it = 0; it < NT; ++it) {
    v8f acc = {};
#pragma unroll
    for (int ct = 0; ct < 8; ++ct) {
      const int c0 = ct * 4 + hi * 2;
      v2f a;
      a.x = q[(head + 4 * c0) * HW + it * 16 + m];
      a.y = q[(head + 4 * (c0 + 1)) * HW + it * 16 + m];
      acc = wmma_f32(a, kb[ct], acc);
    }
    // column (over M=query) reduce: 8 regs in-lane + lane-half exchange
    float tmax = -3.4e38f;
#pragma unroll
    for (int r = 0; r < 8; ++r) tmax = fmaxf(tmax, acc[r] * SCALE);
    tmax = fmaxf(tmax, __shfl_xor(tmax, 16, 32));
    const float mnew = fmaxf(runm, tmax);
    float sl = 0.f;
#pragma unroll
    for (int r = 0; r < 8; ++r) sl += __expf(acc[r] * SCALE - mnew);
    sl += __shfl_xor(sl, 16, 32);
    runs = runs * __expf(runm - mnew) + sl;
    runm = mnew;
  }
  if (hi == 0) {
    ms[head * HW + kcol] = runm;
    ss[head * HW + kcol] = runs;
  }
}

// ---------------- Pass 2: out[c,i] = sum_k exp(a[i,k]-m_k)/s_k * v[c,k] ----------------
__global__ void attn_out_kernel(const float* __restrict__ q, const float* __restrict__ kmat,
                                const float* __restrict__ vt,
                                const float* __restrict__ ms, const float* __restrict__ ss,
                                float* __restrict__ ao) {
  __shared__ float lds[4][16 * 17];        // per-wave 16x16 tile, 17-padded (conflict-free)
  const int lane = threadIdx.x & 31;
  const int wave = threadIdx.x >> 5;
  const int gw = blockIdx.x * 4 + wave;    // 1024 waves total
  const int head = gw >> 8;
  const int itile = gw & 255;
  const int m = lane & 15;
  const int hi = lane >> 4;
  float* tile = lds[wave];

  // hoist Q A-tiles (invariant over key loop)
  v2f qa[8];
#pragma unroll
  for (int ct = 0; ct < 8; ++ct) {
    const int c0 = ct * 4 + hi * 2;
    qa[ct].x = q[(head + 4 * c0) * HW + itile * 16 + m];
    qa[ct].y = q[(head + 4 * (c0 + 1)) * HW + itile * 16 + m];
  }

  v8f out0 = {}, out1 = {};
  for (int kt = 0; kt < NT; ++kt) {
    const int kcol = kt * 16 + m;
    // scores: a(i x k) = Q^T K
    v8f acc = {};
#pragma unroll
    for (int ct = 0; ct < 8; ++ct) {
      const int c0 = ct * 4 + hi * 2;
      v2f b;
      b.x = kmat[(head + 4 * c0) * HW + kcol];
      b.y = kmat[(head + 4 * (c0 + 1)) * HW + kcol];
      acc = wmma_f32(qa[ct], b, acc);
    }
    const float mcol = ms[head * HW + kcol];
    const float iscol = 1.0f / ss[head * HW + kcol];
    // weights, stored transposed [k][i] so the reload matches WMMA B layout
#pragma unroll
    for (int r = 0; r < 8; ++r) {
      const float w = __expf(acc[r] * SCALE - mcol) * iscol;
      tile[m * 17 + r + 8 * hi] = w;       // lane stride 17 -> conflict-free
    }
    // PV: out(c x i) += V(c x k) * W^T(k x i)
#pragma unroll
    for (int kk = 0; kk < 4; ++kk) {
      const int ksub = kk * 4 + hi * 2;
      v2f b;
      b.x = tile[ksub * 17 + m];
      b.y = tile[(ksub + 1) * 17 + m];
      const int kg = kt * 16 + ksub;
      v2f a0, a1;
      a0.x = vt[((size_t)head * HW + kg) * CPH + m];
      a0.y = vt[((size_t)head * HW + kg + 1) * CPH + m];
      a1.x = vt[((size_t)head * HW + kg) * CPH + 16 + m];
      a1.y = vt[((size_t)head * HW + kg + 1) * CPH + 16 + m];
      out0 = wmma_f32(a0, b, out0);
      out1 = wmma_f32(a1, b, out1);
    }
  }
  // D layout: c_idx = ct16*16 + r + 8*hi ; full channel = head + 4*c_idx ; i = itile*16+m
#pragma unroll
  for (int r = 0; r < 8; ++r) {
    const int ci = r + 8 * hi;
    ao[(head + 4 * ci) * HW + itile * 16 + m] = out0[r];
    ao[(head + 4 * (16 + ci)) * HW + itile * 16 + m] = out1[r];
  }
}

extern "C" void kernel_launch(void* const* d_in, const int* in_sizes, int n_in,
                              void* d_out, int out_size, void* d_ws, size_t ws_size,
                              hipStream_t stream) {
  (void)in_sizes; (void)n_in; (void)out_size; (void)ws_size;
  const float* x  = (const float*)d_in[0];
  const float* gg = (const float*)d_in[1];
  const float* gb = (const float*)d_in[2];
  const float* wq = (const float*)d_in[3];
  const float* bq = (const float*)d_in[4];
  const float* wk = (const float*)d_in[5];
  const float* bk = (const float*)d_in[6];
  const float* wv = (const float*)d_in[7];
  const float* bv = (const float*)d_in[8];
  const float* wp = (const float*)d_in[9];
  const float* bp = (const float*)d_in[10];
  float* out = (float*)d_out;

  float* ws = (float*)d_ws;
  float* xn = ws;                    // C*HW
  float* qb = xn + (size_t)C * HW;   // C*HW
  float* kb = qb + (size_t)C * HW;   // C*HW
  float* vt = kb + (size_t)C * HW;   // C*HW (transposed: [head][k][c])
  float* ao = vt + (size_t)C * HW;   // C*HW
  float* ms = ao + (size_t)C * HW;   // HEADS*HW
  float* ss = ms + (size_t)HEADS * HW;
  float* gmean = ss + (size_t)HEADS * HW;  // 32
  float* grstd = gmean + 32;               // 32

  gn_stats_kernel<<<32, 256, 0, stream>>>(x, gmean, grstd);
  gn_apply_kernel<<<(C * HW) / 256, 256, 0, stream>>>(x, gg, gb, gmean, grstd, xn);
  gemm_kernel<<<NT, 256, 0, stream>>>(wq, bq, xn, qb, nullptr, 0);
  gemm_kernel<<<NT, 256, 0, stream>>>(wk, bk, xn, kb, nullptr, 0);
  gemm_kernel<<<NT, 256, 0, stream>>>(wv, bv, xn, vt, nullptr, 1);
  attn_stats_kernel<<<(HEADS * NT) / 8, 256, 0, stream>>>(qb, kb, ms, ss);
  attn_out_kernel<<<(HEADS * NT) / 4, 128, 0, stream>>>(qb, kb, vt, ms, ss, ao);
  gemm_kernel<<<NT, 256, 0, stream>>>(wp, bp, ao, out, x, 2);
}